// cvx_89842125897738
// MI455X (gfx1250) — compile-verified
//
#include <hip/hip_runtime.h>
#include <hip/hip_bf16.h>
#include <stdint.h>

typedef __attribute__((ext_vector_type(2))) float v2f;
typedef __attribute__((ext_vector_type(8))) float v8f;

// ---------------- degree / norm ----------------
__global__ void k_deg_init(float* __restrict__ deg, int N) {
  int i = blockIdx.x * blockDim.x + threadIdx.x;
  if (i < N) deg[i] = 1.0f;  // self loop contributes 1
}

__global__ void k_deg_edges(const int* __restrict__ dst, float* __restrict__ deg, int E) {
  int e = blockIdx.x * blockDim.x + threadIdx.x;
  if (e < E) atomicAdd(&deg[dst[e]], 1.0f);
}

__global__ void k_rsqrt(float* __restrict__ d, int N) {
  int i = blockIdx.x * blockDim.x + threadIdx.x;
  if (i < N) d[i] = rsqrtf(d[i]);
}

// ---------------- WMMA GEMM: C[N x O] = A[N x K] * B[K x O] (+bias, relu) ----------------
// One wave computes one 16x16 tile of C via V_WMMA_F32_16X16X4_F32, K stepped by 4.
// blockDim.x = (O/16)*32 waves-per-row-tile; gridDim.x = ceil(N/16).
// A-matrix lane layout (32-bit A 16x4): lane L holds row M=L%16, VGPR0: K=2*(L>=16), VGPR1: K=+1.
// C/D layout: VGPR r, lanes 0-15 -> M=r, N=lane; lanes 16-31 -> M=8+r, N=lane-16.
__global__ void k_gemm_wmma(const float* __restrict__ A, const float* __restrict__ B,
                            const float* __restrict__ bias, float* __restrict__ C,
                            int Nrows, int K, int O, int relu) {
  const int lane = threadIdx.x & 31;
  const int wv   = threadIdx.x >> 5;
  const int m0   = blockIdx.x * 16;
  const int n0   = wv * 16;
  const int half = lane >> 4;   // 0: lanes 0-15, 1: lanes 16-31
  const int lm   = lane & 15;

  int arow = m0 + lm;
  if (arow >= Nrows) arow = Nrows - 1;           // clamp (no divergence; EXEC stays full)
  const float* __restrict__ Ap = A + (long long)arow * K;
  const float* __restrict__ Bp = B + n0 + lm;

  v8f c = {};
  for (int k0 = 0; k0 < K; k0 += 4) {
    const int ka = k0 + 2 * half;
    v2f a, b;
    a.x = Ap[ka];
    a.y = Ap[ka + 1];
    b.x = Bp[(long long)ka * O];
    b.y = Bp[(long long)(ka + 1) * O];
    c = __builtin_amdgcn_wmma_f32_16x16x4_f32(false, a, false, b, (short)0, c, false, false);
  }

  const int col = n0 + lm;
  const float bv = bias ? bias[col] : 0.0f;
#pragma unroll
  for (int r = 0; r < 8; ++r) {
    int row = m0 + r + half * 8;
    if (row < Nrows) {
      float v = c[r] + bv;
      if (relu) v = fmaxf(v, 0.0f);
      C[(long long)row * O + col] = v;
    }
  }
}

// ---------------- aggregation ----------------
// out[n, :] = hw[n, :] * dinv[n]^2   (self-loop term)
__global__ void k_self_init(const float* __restrict__ hw, const float* __restrict__ dinv,
                            float* __restrict__ out, int N, int D) {
  long long i = (long long)blockIdx.x * blockDim.x + threadIdx.x;
  long long total = (long long)N * D;
  if (i < total) {
    int n = (int)(i / D);
    float w = dinv[n];
    out[i] = hw[i] * (w * w);
  }
}

// one wave per edge (grid-stride): out[dst,:] += hw[src,:] * dinv[src]*dinv[dst]
template <int D>
__global__ void k_agg_edges(const int* __restrict__ src, const int* __restrict__ dst,
                            const float* __restrict__ dinv, const float* __restrict__ hw,
                            float* __restrict__ out, int E) {
  const int lane = threadIdx.x & 31;
  long long w  = (long long)blockIdx.x * (blockDim.x >> 5) + (threadIdx.x >> 5);
  long long nw = (long long)gridDim.x * (blockDim.x >> 5);
  for (long long e = w; e < E; e += nw) {
    const int s = src[e];
    const int d = dst[e];
    const float nf = dinv[s] * dinv[d];
    const float* __restrict__ hs = hw + (long long)s * D;
    float* __restrict__ od = out + (long long)d * D;
#pragma unroll
    for (int c0 = 0; c0 < D; c0 += 32) {
      atomicAdd(&od[c0 + lane], hs[c0 + lane] * nf);
    }
  }
}

// h[i] = relu(h[i] + b[i % D]); D is a power of two
__global__ void k_bias_relu(float* __restrict__ h, const float* __restrict__ b, int N, int D) {
  long long i = (long long)blockIdx.x * blockDim.x + threadIdx.x;
  long long total = (long long)N * D;
  if (i < total) h[i] = fmaxf(h[i] + b[(int)(i & (D - 1))], 0.0f);
}

// ---------------- heads ----------------
// per node: s_src = h.Wsw[0:64], s_dst = h.Wsw[64:128], voltage output
__global__ void k_node_head(const float* __restrict__ h, const float* __restrict__ Wsw,
                            const float* __restrict__ Wv, const float* __restrict__ bv,
                            float* __restrict__ s_src, float* __restrict__ s_dst,
                            float* __restrict__ out_volt, int N) {
  int n = blockIdx.x * blockDim.x + threadIdx.x;
  if (n >= N) return;
  const float* __restrict__ hn = h + (long long)n * 64;
  float a0 = 0.f, a1 = 0.f, av = 0.f;
#pragma unroll
  for (int i = 0; i < 64; ++i) {
    const float v = hn[i];
    a0 += v * Wsw[i];
    a1 += v * Wsw[64 + i];
    av += v * Wv[i];
  }
  s_src[n] = a0;
  s_dst[n] = a1;
  const float vr = 1.0f / (1.0f + __expf(-(av + bv[0])));
  float vq = 0.9f + 0.2f * vr;
  float vw = vq * vq;
  vw = fminf(fmaxf(vw, 0.81f), 1.21f);
  out_volt[n] = sqrtf(vw);
}

__global__ void k_edge_head(const int* __restrict__ src, const int* __restrict__ dst,
                            const float* __restrict__ s_src, const float* __restrict__ s_dst,
                            const float* __restrict__ bsw, float* __restrict__ out, int E) {
  int e = blockIdx.x * blockDim.x + threadIdx.x;
  if (e >= E) return;
  const float z = s_src[src[e]] + s_dst[dst[e]] + bsw[0];
  float yw = 1.0f / (1.0f + __expf(-z));
  yw = fminf(fmaxf(yw, 0.0f), 1.0f);
  out[e] = yw;
}

extern "C" void kernel_launch(void* const* d_in, const int* in_sizes, int n_in,
                              void* d_out, int out_size, void* d_ws, size_t ws_size,
                              hipStream_t stream) {
  const float* x    = (const float*)d_in[0];
  const int*   ei   = (const int*)d_in[1];
  const float* Wenc = (const float*)d_in[2];
  const float* benc = (const float*)d_in[3];
  const float* Wg0  = (const float*)d_in[4];
  const float* bg0  = (const float*)d_in[5];
  const float* Wg1  = (const float*)d_in[6];
  const float* bg1  = (const float*)d_in[7];
  const float* Wsw  = (const float*)d_in[8];
  const float* bsw  = (const float*)d_in[9];
  const float* Wv   = (const float*)d_in[10];
  const float* bv   = (const float*)d_in[11];

  const int N = in_sizes[0] / 16;   // F_IN = 16
  const int E = in_sizes[1] / 2;    // edge_index is [2, E]
  const int* src = ei;
  const int* dst = ei + E;

  float* ws   = (float*)d_ws;
  float* dinv = ws;                          // N
  float* bufA = dinv + N;                    // N*128 : h0 -> out0/h1 -> out1/h
  float* bufB = bufA + (long long)N * 128;   // N*128 : hw0 -> hw1 -> s_src/s_dst

  float* out_y = (float*)d_out;   // [E] switch scores
  float* out_v = out_y + E;       // [N] voltage scores

  const int T = 256;
  const int Mt = (N + 15) / 16;
  const int edge_wave_blocks = (E + 7) / 8;   // 8 waves per 256-thread block

  // 1. degrees -> dinv
  k_deg_init<<<(N + T - 1) / T, T, 0, stream>>>(dinv, N);
  k_deg_edges<<<(E + T - 1) / T, T, 0, stream>>>(dst, dinv, E);
  k_rsqrt<<<(N + T - 1) / T, T, 0, stream>>>(dinv, N);

  // 2. encoder: h0 = relu(x @ Wenc + benc)   [N,128] -> bufA
  k_gemm_wmma<<<Mt, 256, 0, stream>>>(x, Wenc, benc, bufA, N, 16, 128, 1);

  // 3. GCN layer 0
  k_gemm_wmma<<<Mt, 256, 0, stream>>>(bufA, Wg0, nullptr, bufB, N, 128, 128, 0);  // hw0
  {
    long long tot = (long long)N * 128;
    k_self_init<<<(int)((tot + T - 1) / T), T, 0, stream>>>(bufB, dinv, bufA, N, 128);
    k_agg_edges<128><<<edge_wave_blocks, 256, 0, stream>>>(src, dst, dinv, bufB, bufA, E);
    k_bias_relu<<<(int)((tot + T - 1) / T), T, 0, stream>>>(bufA, bg0, N, 128);
  }

  // 4. GCN layer 1
  k_gemm_wmma<<<Mt, 128, 0, stream>>>(bufA, Wg1, nullptr, bufB, N, 128, 64, 0);   // hw1
  {
    long long tot = (long long)N * 64;
    k_self_init<<<(int)((tot + T - 1) / T), T, 0, stream>>>(bufB, dinv, bufA, N, 64);
    k_agg_edges<64><<<edge_wave_blocks, 256, 0, stream>>>(src, dst, dinv, bufB, bufA, E);
    k_bias_relu<<<(int)((tot + T - 1) / T), T, 0, stream>>>(bufA, bg1, N, 64);
  }

  // 5. heads: precompute per-node partial dots, then cheap per-edge combine
  float* s_src = bufB;        // hw1 dead now
  float* s_dst = bufB + N;
  k_node_head<<<(N + T - 1) / T, T, 0, stream>>>(bufA, Wsw, Wv, bv, s_src, s_dst, out_v, N);
  k_edge_head<<<(E + T - 1) / T, T, 0, stream>>>(src, dst, s_src, s_dst, bsw, out_y, E);
}